// ExchangeableLayer_63170378989803
// MI455X (gfx1250) — compile-verified
//
#include <hip/hip_runtime.h>

#define U 64
#define SEG_EPS 1e-10f
#define MDIM 10000
#define NDIM 50000

typedef float v2f __attribute__((ext_vector_type(2)));
typedef float v8f __attribute__((ext_vector_type(8)));

// ---------------- workspace layout (float offsets) ----------------
#define OFF_COLSUM0 0                       // M*64
#define OFF_COLCNT0 (OFF_COLSUM0 + MDIM*U)  // M
#define OFF_ROWSUM0 (OFF_COLCNT0 + MDIM)    // N*64
#define OFF_ROWCNT0 (OFF_ROWSUM0 + NDIM*U)  // N
#define OFF_COLSUM1 (OFF_ROWCNT0 + NDIM)    // M*64
#define OFF_COLCNT1 (OFF_COLSUM1 + MDIM*U)  // M
#define OFF_ROWSUM2 (OFF_COLCNT1 + MDIM)    // N*64
#define OFF_ROWCNT2 (OFF_ROWSUM2 + NDIM*U)  // N
#define OFF_TOT0    (OFF_ROWCNT2 + NDIM)    // 64
#define OFF_TOT1    (OFF_TOT0 + U)          // 64
#define OFF_TOT2    (OFF_TOT1 + U)          // 64
#define OFF_BIAS    (OFF_TOT2 + U)          // 64
#define ZERO_FLOATS (OFF_BIAS + U)          // everything above gets zeroed
#define OFF_COLFEAT (ZERO_FLOATS)           // M*64 (fully overwritten)
#define OFF_ROWFEAT (OFF_COLFEAT + MDIM*U)  // N*64 (fully overwritten)

// ---------------- zero the accumulator region ----------------
__global__ void zero_ws_kernel(float4* __restrict__ p, int n4) {
    int i = blockIdx.x * blockDim.x + threadIdx.x;
    if (i < n4) p[i] = make_float4(0.f, 0.f, 0.f, 0.f);
}

// ---------------- segment-sum scatters (L2-resident atomics) ----------------
__global__ void scatter_dual_kernel(const float* __restrict__ vals,
                                    const int* __restrict__ rows,
                                    const int* __restrict__ cols,
                                    float* __restrict__ rowsum, float* __restrict__ rowcnt,
                                    float* __restrict__ colsum, float* __restrict__ colcnt,
                                    int total) {
    int gid = blockIdx.x * blockDim.x + threadIdx.x;
    if (gid >= total) return;
    int e = gid >> 6, j = gid & 63;
    float v = vals[gid];
    int r = rows[e], c = cols[e];
    atomicAdd(&rowsum[r * U + j], v);
    atomicAdd(&colsum[c * U + j], v);
    if (j == 0) {
        atomicAdd(&rowcnt[r], 1.0f);
        atomicAdd(&colcnt[c], 1.0f);
    }
}

__global__ void scatter_one_kernel(const float* __restrict__ vals,
                                   const int* __restrict__ ids,
                                   float* __restrict__ sum, float* __restrict__ cnt,
                                   int total) {
    int gid = blockIdx.x * blockDim.x + threadIdx.x;
    if (gid >= total) return;
    int e = gid >> 6, j = gid & 63;
    float v = vals[gid];
    int s = ids[e];
    atomicAdd(&sum[s * U + j], v);
    if (j == 0) atomicAdd(&cnt[s], 1.0f);
}

// ---------------- reduce per-segment sums -> 64-wide totals ----------------
__global__ void col_reduce_kernel(const float* __restrict__ s, float* __restrict__ tot, int R) {
    int j = threadIdx.x & 63;
    int rowsPerBlock = blockDim.x >> 6;
    int r = blockIdx.x * rowsPerBlock + (threadIdx.x >> 6);
    int stride = gridDim.x * rowsPerBlock;
    float acc = 0.f;
    for (; r < R; r += stride) acc += s[r * U + j];
    atomicAdd(&tot[j], acc);
}

// ---------------- bias64 = Σ mean_k * theta[k][j] (+theta_b) ----------------
__global__ void bias_kernel(const float* __restrict__ tot0, const float* __restrict__ tot1,
                            const float* __restrict__ tot2,
                            const float* __restrict__ th11, const float* __restrict__ th1x011,
                            const float* __restrict__ th2x011, const float* __restrict__ thb,
                            float* __restrict__ bias, float inv0, float inv1, float inv2) {
    int j = threadIdx.x;
    if (j >= U) return;
    float acc = thb[j];
    for (int k = 0; k < U; ++k) {
        acc += tot0[k] * inv0 * th11[k * U + j];
        acc += tot1[k] * inv1 * th1x011[k * U + j];
        acc += tot2[k] * inv2 * th2x011[k * U + j];
    }
    bias[j] = acc;
}

// ---------------- feat[R,64] = (sumA/cntA)@thA + (sumB/cntB)@thB via fp32 WMMA ----------------
__global__ void feat_gemm_kernel(const float* __restrict__ sumA, const float* __restrict__ cntA,
                                 const float* __restrict__ thA,
                                 const float* __restrict__ sumB, const float* __restrict__ cntB,
                                 const float* __restrict__ thB,
                                 float* __restrict__ feat, int numTiles) {
    __shared__ float sA[U * U];
    __shared__ float sB[U * U];
    for (int i = threadIdx.x; i < U * U; i += blockDim.x) { sA[i] = thA[i]; sB[i] = thB[i]; }
    __syncthreads();

    int tile = blockIdx.x * (blockDim.x >> 5) + (threadIdx.x >> 5);
    if (tile >= numTiles) return;               // wave-uniform: EXEC stays all-ones
    int lane = threadIdx.x & 31;
    int m15 = lane & 15, half = lane >> 4;
    int row = tile * 16 + m15;
    float invA = 1.0f / (cntA[row] + SEG_EPS);
    float invB = 1.0f / (cntB[row] + SEG_EPS);
    const float* aRow = sumA + row * U;
    const float* bRow = sumB + row * U;

    for (int ct = 0; ct < 4; ++ct) {
        v8f c = {0.f, 0.f, 0.f, 0.f, 0.f, 0.f, 0.f, 0.f};
        int n = ct * 16 + m15;
#pragma unroll
        for (int kk = 0; kk < 16; ++kk) {       // A-matrix contribution, K = 64
            int k0 = kk * 4 + 2 * half;         // A/B 16x4 f32 layout: K = vgpr + 2*half
            v2f a; a.x = aRow[k0] * invA; a.y = aRow[k0 + 1] * invA;
            v2f b; b.x = sA[k0 * U + n]; b.y = sA[(k0 + 1) * U + n];
            c = __builtin_amdgcn_wmma_f32_16x16x4_f32(false, a, false, b, (short)0, c, false, false);
        }
#pragma unroll
        for (int kk = 0; kk < 16; ++kk) {       // B-matrix contribution accumulates into same C
            int k0 = kk * 4 + 2 * half;
            v2f a; a.x = bRow[k0] * invB; a.y = bRow[k0 + 1] * invB;
            v2f b; b.x = sB[k0 * U + n]; b.y = sB[(k0 + 1) * U + n];
            c = __builtin_amdgcn_wmma_f32_16x16x4_f32(false, a, false, b, (short)0, c, false, false);
        }
#pragma unroll
        for (int r = 0; r < 8; ++r)             // C/D layout: M = r + 8*half, N = lane&15
            feat[(tile * 16 + r + 8 * half) * U + n] = c[r];
    }
}

// ---------------- out[e] = relu(t0_values[e]@th00 + colfeat[col] + rowfeat[row] + bias) ----------------
__global__ void main_gemm_kernel(const float* __restrict__ t0v, const float* __restrict__ th00,
                                 const int* __restrict__ rows, const int* __restrict__ cols,
                                 const float* __restrict__ colfeat, const float* __restrict__ rowfeat,
                                 const float* __restrict__ bias,
                                 float* __restrict__ out, int numTiles) {
    __shared__ float sTh[U * U];
    __shared__ float sBias[U];
    for (int i = threadIdx.x; i < U * U; i += blockDim.x) sTh[i] = th00[i];
    if (threadIdx.x < U) sBias[threadIdx.x] = bias[threadIdx.x];
    __syncthreads();

    int tile = blockIdx.x * (blockDim.x >> 5) + (threadIdx.x >> 5);
    if (tile >= numTiles) return;               // wave-uniform guard
    int lane = threadIdx.x & 31;
    int m15 = lane & 15, half = lane >> 4;
    int ebase = tile * 16;
    const float* aRow = t0v + (ebase + m15) * U;

    v8f acc[4];
#pragma unroll
    for (int ct = 0; ct < 4; ++ct) acc[ct] = (v8f){0.f, 0.f, 0.f, 0.f, 0.f, 0.f, 0.f, 0.f};

#pragma unroll
    for (int kk = 0; kk < 16; ++kk) {           // load A once per k-step, reuse across 4 N-tiles
        int k0 = kk * 4 + 2 * half;
        v2f a; a.x = aRow[k0]; a.y = aRow[k0 + 1];
#pragma unroll
        for (int ct = 0; ct < 4; ++ct) {
            int n = ct * 16 + m15;
            v2f b; b.x = sTh[k0 * U + n]; b.y = sTh[(k0 + 1) * U + n];
            acc[ct] = __builtin_amdgcn_wmma_f32_16x16x4_f32(false, a, false, b, (short)0, acc[ct], false, false);
        }
    }

#pragma unroll
    for (int r = 0; r < 8; ++r) {
        int e2 = ebase + r + 8 * half;
        int cb = cols[e2] * U;
        int rb = rows[e2] * U;
        int ob = e2 * U;
#pragma unroll
        for (int ct = 0; ct < 4; ++ct) {
            int n = ct * 16 + m15;
            float v = acc[ct][r] + colfeat[cb + n] + rowfeat[rb + n] + sBias[n];
            out[ob + n] = fmaxf(v, 0.0f);
        }
    }
}

// ---------------- host ----------------
extern "C" void kernel_launch(void* const* d_in, const int* in_sizes, int n_in,
                              void* d_out, int out_size, void* d_ws, size_t ws_size,
                              hipStream_t stream) {
    const float* t0v      = (const float*)d_in[0];
    const float* t1v      = (const float*)d_in[1];
    const float* t2v      = (const float*)d_in[2];
    const float* th00     = (const float*)d_in[3];
    const float* th10     = (const float*)d_in[4];
    const float* th01     = (const float*)d_in[5];
    const float* th11     = (const float*)d_in[6];
    const float* th1x0_10 = (const float*)d_in[7];
    const float* th1x0_11 = (const float*)d_in[8];
    const float* th2x0_01 = (const float*)d_in[9];
    const float* th2x0_11 = (const float*)d_in[10];
    const float* thb      = (const float*)d_in[11];
    const int* t0_rows    = (const int*)d_in[12];
    const int* t0_cols    = (const int*)d_in[13];
    const int* t1_cols    = (const int*)d_in[14];
    const int* t2_rows    = (const int*)d_in[15];

    const int NNZ0 = in_sizes[0] / U;
    const int NNZ1 = in_sizes[1] / U;
    const int NNZ2 = in_sizes[2] / U;

    float* ws = (float*)d_ws;
    float* colsum0 = ws + OFF_COLSUM0;  float* colcnt0 = ws + OFF_COLCNT0;
    float* rowsum0 = ws + OFF_ROWSUM0;  float* rowcnt0 = ws + OFF_ROWCNT0;
    float* colsum1 = ws + OFF_COLSUM1;  float* colcnt1 = ws + OFF_COLCNT1;
    float* rowsum2 = ws + OFF_ROWSUM2;  float* rowcnt2 = ws + OFF_ROWCNT2;
    float* tot0 = ws + OFF_TOT0; float* tot1 = ws + OFF_TOT1; float* tot2 = ws + OFF_TOT2;
    float* biasb = ws + OFF_BIAS;
    float* colfeat = ws + OFF_COLFEAT;
    float* rowfeat = ws + OFF_ROWFEAT;

    // 1) zero all accumulators (float4-vectorized; ZERO_FLOATS is a multiple of 4)
    {
        int n4 = ZERO_FLOATS / 4;
        zero_ws_kernel<<<(n4 + 255) / 256, 256, 0, stream>>>((float4*)d_ws, n4);
    }
    // 2) segment sums via f32 atomics (accumulators fit in 192MB L2)
    {
        int tot = NNZ0 * U;
        scatter_dual_kernel<<<(tot + 255) / 256, 256, 0, stream>>>(
            t0v, t0_rows, t0_cols, rowsum0, rowcnt0, colsum0, colcnt0, tot);
    }
    {
        int tot = NNZ1 * U;
        scatter_one_kernel<<<(tot + 255) / 256, 256, 0, stream>>>(t1v, t1_cols, colsum1, colcnt1, tot);
    }
    {
        int tot = NNZ2 * U;
        scatter_one_kernel<<<(tot + 255) / 256, 256, 0, stream>>>(t2v, t2_rows, rowsum2, rowcnt2, tot);
    }
    // 3) totals (total sum = sum of segment sums -> exact jnp.mean semantics after /NNZ)
    col_reduce_kernel<<<128, 256, 0, stream>>>(colsum0, tot0, MDIM);
    col_reduce_kernel<<<128, 256, 0, stream>>>(colsum1, tot1, MDIM);
    col_reduce_kernel<<<256, 256, 0, stream>>>(rowsum2, tot2, NDIM);
    // 4) fused bias vector
    bias_kernel<<<1, 64, 0, stream>>>(tot0, tot1, tot2, th11, th1x0_11, th2x0_11, thb, biasb,
                                      1.0f / (float)NNZ0, 1.0f / (float)NNZ1, 1.0f / (float)NNZ2);
    // 5) per-column / per-row feature GEMMs (fp32 WMMA)
    {
        int tiles = MDIM / 16;
        feat_gemm_kernel<<<(tiles + 7) / 8, 256, 0, stream>>>(
            colsum0, colcnt0, th10, colsum1, colcnt1, th1x0_10, colfeat, tiles);
    }
    {
        int tiles = NDIM / 16;
        feat_gemm_kernel<<<(tiles + 7) / 8, 256, 0, stream>>>(
            rowsum0, rowcnt0, th01, rowsum2, rowcnt2, th2x0_01, rowfeat, tiles);
    }
    // 6) main GEMM + gathered epilogue + relu
    {
        int tiles = NNZ0 / 16;
        main_gemm_kernel<<<(tiles + 7) / 8, 256, 0, stream>>>(
            t0v, th00, t0_rows, t0_cols, colfeat, rowfeat, biasb, (float*)d_out, tiles);
    }
}